// TopKGate_47579647705877
// MI455X (gfx1250) — compile-verified
//
#include <hip/hip_runtime.h>
#include <hip/hip_bf16.h>
#include <math.h>

// ---------------------------------------------------------------------------
// TopKGate (MoE top-2 gating) for MI455X / gfx1250, wave32.
//   logits = x @ w^T  : M=8192, N=64, K=1024, fp32 -> V_WMMA_F32_16X16X4_F32
//   outputs: l_aux[1], combine[8192,64,256], dispatch[8192,64,256], counts[64]
// ---------------------------------------------------------------------------

#define S_TOKENS 8192
#define MODEL_DIM 1024
#define NUM_EXPERTS 64
#define CAPACITY 256
#define CE_ELEMS (134217728LL) // 8192*64*256

typedef float v2f __attribute__((ext_vector_type(2)));
typedef float v8f __attribute__((ext_vector_type(8)));

// ---------------------------------------------------------------------------
// 0) zero-fill d_out (float4 bulk + tail) and the me_sum accumulator
// ---------------------------------------------------------------------------
__global__ __launch_bounds__(256)
void moe_zero_out(float4* __restrict__ p4, long long n4,
                  float* __restrict__ tail, int ntail,
                  float* __restrict__ me_sum) {
    long long tid = (long long)blockIdx.x * blockDim.x + threadIdx.x;
    long long stride = (long long)gridDim.x * blockDim.x;
    float4 z = make_float4(0.f, 0.f, 0.f, 0.f);
    for (long long i = tid; i < n4; i += stride) p4[i] = z;
    if (tid < ntail) tail[tid] = 0.f;
    if (tid < NUM_EXPERTS) me_sum[tid] = 0.f;
}

// ---------------------------------------------------------------------------
// 1) logits = x @ w^T via WMMA fp32 16x16x4.
//    One wave = 16 token-rows x all 64 experts (4 accumulators).
//    A layout (16x4 f32): v0 = {lane0-15: K=0, lane16-31: K=2}, v1 = {K=1, K=3}
//      -> lane (r,h) loads x[m0+r][k0+2h .. +1]    (8B-aligned b64 load)
//    B layout (4x16 f32): v0 = {lane0-15: K=0 row, lane16-31: K=2 row}
//      -> lane (r,h) loads w[n0+r][k0+2h .. +1]
//    C/D layout: acc[i] = D[m0 + i + 8h][n0 + r]
// ---------------------------------------------------------------------------
__global__ __launch_bounds__(256)
void moe_logits_wmma(const float* __restrict__ x, const float* __restrict__ w,
                     float* __restrict__ logits) {
    const int lane = threadIdx.x & 31;
    const int wave = threadIdx.x >> 5;
    const int m0 = (blockIdx.x * 8 + wave) * 16;
    const int r = lane & 15;
    const int h = lane >> 4;

    const float* xr = x + (size_t)(m0 + r) * MODEL_DIM + 2 * h;
    const float* w0 = w + (size_t)(0  + r) * MODEL_DIM + 2 * h;
    const float* w1 = w + (size_t)(16 + r) * MODEL_DIM + 2 * h;
    const float* w2 = w + (size_t)(32 + r) * MODEL_DIM + 2 * h;
    const float* w3 = w + (size_t)(48 + r) * MODEL_DIM + 2 * h;

    v8f acc0 = {}, acc1 = {}, acc2 = {}, acc3 = {};

#pragma unroll 4
    for (int k0 = 0; k0 < MODEL_DIM; k0 += 4) {
        v2f a  = *(const v2f*)(xr + k0);
        v2f b0 = *(const v2f*)(w0 + k0);
        v2f b1 = *(const v2f*)(w1 + k0);
        v2f b2 = *(const v2f*)(w2 + k0);
        v2f b3 = *(const v2f*)(w3 + k0);
        acc0 = __builtin_amdgcn_wmma_f32_16x16x4_f32(false, a, false, b0,
                                                     (short)0, acc0, false, false);
        acc1 = __builtin_amdgcn_wmma_f32_16x16x4_f32(false, a, false, b1,
                                                     (short)0, acc1, false, false);
        acc2 = __builtin_amdgcn_wmma_f32_16x16x4_f32(false, a, false, b2,
                                                     (short)0, acc2, false, false);
        acc3 = __builtin_amdgcn_wmma_f32_16x16x4_f32(false, a, false, b3,
                                                     (short)0, acc3, false, false);
    }

#pragma unroll
    for (int i = 0; i < 8; ++i) {
        int row = m0 + i + 8 * h;
        float* dst = logits + (size_t)row * NUM_EXPERTS + r;
        dst[0]  = acc0[i];
        dst[16] = acc1[i];
        dst[32] = acc2[i];
        dst[48] = acc3[i];
    }
}

// ---------------------------------------------------------------------------
// 2) per-token gating: softmax, top1 (argmax logits), top2 over logits+gumbel
//    with winner masked to -inf. Accumulates sum_s gates[s][e] via LDS atomics.
// ---------------------------------------------------------------------------
__global__ __launch_bounds__(256)
void moe_gate_token(const float* __restrict__ logits,
                    const float* __restrict__ gumbel,
                    int* __restrict__ idx1, int* __restrict__ idx2,
                    float* __restrict__ g1raw, float* __restrict__ g2raw,
                    float* __restrict__ me_sum) {
    __shared__ float sm[NUM_EXPERTS];
    if (threadIdx.x < NUM_EXPERTS) sm[threadIdx.x] = 0.f;
    __syncthreads();

    int s = blockIdx.x * blockDim.x + threadIdx.x;
    const float* lrow = logits + (size_t)s * NUM_EXPERTS;
    const float* grow = gumbel + (size_t)s * NUM_EXPERTS;

    // pass 1: max + first-occurrence argmax (matches jnp.argmax)
    float mx = lrow[0];
    int i1 = 0;
    for (int e = 1; e < NUM_EXPERTS; ++e) {
        float v = lrow[e];
        if (v > mx) { mx = v; i1 = e; }
    }
    // pass 2: softmax denominator
    float den = 0.f;
    for (int e = 0; e < NUM_EXPERTS; ++e) den += expf(lrow[e] - mx);
    float invden = 1.f / den;

    // pass 3: per-expert gate accumulation + second argmax
    float best2 = -INFINITY;
    int i2 = 0;
    for (int e = 0; e < NUM_EXPERTS; ++e) {
        float ge = expf(lrow[e] - mx) * invden;
        atomicAdd(&sm[e], ge);  // LDS ds_add_f32
        float v2 = (e == i1) ? -INFINITY : (lrow[e] + grow[e]);
        if (v2 > best2) { best2 = v2; i2 = e; }
    }

    idx1[s]  = i1;
    idx2[s]  = i2;
    g1raw[s] = expf(lrow[i1] - mx) * invden;
    g2raw[s] = expf(lrow[i2] - mx) * invden;

    __syncthreads();
    if (threadIdx.x < NUM_EXPERTS) atomicAdd(&me_sum[threadIdx.x], sm[threadIdx.x]);
}

// ---------------------------------------------------------------------------
// 3) sequential rank scan (cumsum semantics), one block of 64 threads.
//    locations1_s = rank among tokens with idx1==e
//    locations2_s = rank among tokens with idx2==e + total_count1[e]
//    Also finalizes l_aux = 64 * sum_e me[e]*ce[e] and exp_counts.
// ---------------------------------------------------------------------------
__global__ __launch_bounds__(64)
void moe_scan(const int* __restrict__ idx1, const int* __restrict__ idx2,
              const float* __restrict__ me_sum,
              int* __restrict__ loc1, int* __restrict__ loc2,
              float* __restrict__ out_laux, float* __restrict__ out_counts) {
    __shared__ int cnt1sh[NUM_EXPERTS];
    int e = threadIdx.x;

    int c1 = 0;
    for (int s = 0; s < S_TOKENS; ++s)
        if (idx1[s] == e) { loc1[s] = c1; ++c1; }
    cnt1sh[e] = c1;
    out_counts[e] = (float)c1;  // exp_counts (pre-capacity), as float

    int c2 = 0;
    for (int s = 0; s < S_TOKENS; ++s)
        if (idx2[s] == e) { loc2[s] = c2 + c1; ++c2; }

    __syncthreads();
    if (e == 0) {
        float acc = 0.f;
        for (int i = 0; i < NUM_EXPERTS; ++i) {
            float me = me_sum[i] * (1.f / (float)S_TOKENS);
            float ce = (float)cnt1sh[i] * (1.f / (float)S_TOKENS);
            acc += me * ce;
        }
        // mean over e of me*ce, times e*e  ->  (sum/64)*4096 = 64*sum
        out_laux[0] = acc * 64.f;
    }
}

// ---------------------------------------------------------------------------
// 4) scatter: capacity-drop, renormalize, write <=2 entries per token into
//    combine_weights and dispatch_mask (already zero-filled).
// ---------------------------------------------------------------------------
__global__ __launch_bounds__(256)
void moe_scatter(const int* __restrict__ idx1, const int* __restrict__ idx2,
                 const int* __restrict__ loc1, const int* __restrict__ loc2,
                 const float* __restrict__ g1raw, const float* __restrict__ g2raw,
                 float* __restrict__ combine, float* __restrict__ dispatch) {
    int s = blockIdx.x * blockDim.x + threadIdx.x;
    if (s >= S_TOKENS) return;

    int i1 = idx1[s], i2 = idx2[s];
    int l1 = loc1[s], l2 = loc2[s];
    bool keep1 = (l1 < CAPACITY);
    bool keep2 = (l2 < CAPACITY);

    float g1 = keep1 ? g1raw[s] : 0.f;
    float g2 = keep2 ? g2raw[s] : 0.f;
    float den = fmaxf(g1 + g2, 1.1920929e-07f);  // FLT_EPSILON
    g1 /= den;
    g2 /= den;

    size_t base = (size_t)s * NUM_EXPERTS * CAPACITY;
    if (keep1) {
        size_t o = base + (size_t)i1 * CAPACITY + (size_t)l1;
        combine[o]  = g1;
        dispatch[o] = (g1 != 0.f) ? 1.f : 0.f;
    }
    if (keep2) {
        size_t o = base + (size_t)i2 * CAPACITY + (size_t)l2;
        combine[o]  = g2;
        dispatch[o] = (g2 != 0.f) ? 1.f : 0.f;
    }
}

// ---------------------------------------------------------------------------
// launcher
// ---------------------------------------------------------------------------
extern "C" void kernel_launch(void* const* d_in, const int* in_sizes, int n_in,
                              void* d_out, int out_size, void* d_ws, size_t ws_size,
                              hipStream_t stream) {
    const float* x      = (const float*)d_in[0];
    const float* w      = (const float*)d_in[1];
    const float* gumbel = (const float*)d_in[2];

    float* out = (float*)d_out;
    // output layout (return order): l_aux[1], combine[CE], dispatch[CE], counts[64]
    float* out_laux     = out;
    float* out_combine  = out + 1;
    float* out_dispatch = out + 1 + (size_t)CE_ELEMS;
    float* out_counts   = out + 1 + 2 * (size_t)CE_ELEMS;

    // workspace layout
    char* ws = (char*)d_ws;
    float* logits = (float*)ws;                                   // 8192*64 f32 = 2 MB
    size_t off = (size_t)S_TOKENS * NUM_EXPERTS * sizeof(float);
    int*   idx1   = (int*)(ws + off);   off += S_TOKENS * sizeof(int);
    int*   idx2   = (int*)(ws + off);   off += S_TOKENS * sizeof(int);
    int*   loc1   = (int*)(ws + off);   off += S_TOKENS * sizeof(int);
    int*   loc2   = (int*)(ws + off);   off += S_TOKENS * sizeof(int);
    float* g1raw  = (float*)(ws + off); off += S_TOKENS * sizeof(float);
    float* g2raw  = (float*)(ws + off); off += S_TOKENS * sizeof(float);
    float* me_sum = (float*)(ws + off); off += NUM_EXPERTS * sizeof(float);

    // 0) zero the whole output region (bulk float4 + scalar tail) and me_sum
    long long total = 1LL + 2LL * CE_ELEMS + NUM_EXPERTS;
    long long n4 = total / 4;
    int ntail = (int)(total - 4 * n4);
    float* tail = out + 4 * n4;
    moe_zero_out<<<32768, 256, 0, stream>>>((float4*)out, n4, tail, ntail, me_sum);

    // 1) logits GEMM: 512 M-tiles of 16 rows, 8 waves per 256-thread block
    moe_logits_wmma<<<S_TOKENS / 16 / 8, 256, 0, stream>>>(x, w, logits);

    // 2) per-token top-2 gating
    moe_gate_token<<<S_TOKENS / 256, 256, 0, stream>>>(logits, gumbel, idx1, idx2,
                                                       g1raw, g2raw, me_sum);

    // 3) per-expert sequential rank scan + l_aux + exp_counts
    moe_scan<<<1, 64, 0, stream>>>(idx1, idx2, me_sum, loc1, loc2,
                                   out_laux, out_counts);

    // 4) sparse scatter into combine/dispatch
    moe_scatter<<<S_TOKENS / 256, 256, 0, stream>>>(idx1, idx2, loc1, loc2,
                                                    g1raw, g2raw,
                                                    out_combine, out_dispatch);
}